// SemanticSimilarity_56229711839979
// MI455X (gfx1250) — compile-verified
//
#include <hip/hip_runtime.h>
#include <hip/hip_bf16.h>

typedef float v2f __attribute__((ext_vector_type(2)));
typedef float v8f __attribute__((ext_vector_type(8)));

#define NSP 32                 // superpixels
#define NCH 64                 // channels
#define HW  (512*512)          // pixels per image
#define NB  16                 // batch
#define BPB 64                 // accumulation blocks per batch
#define PIX_PER_BLK (HW/BPB)   // 4096 pixels per block
#define THREADS 256

// ---------------------------------------------------------------------------
// Kernel 0: zero the workspace (sums [NB*NSP*NCH] + counts [NB*NSP]).
// ---------------------------------------------------------------------------
__global__ __launch_bounds__(256) void zero_ws(float* __restrict__ ws, int n) {
  int i = blockIdx.x * 256 + threadIdx.x;
  if (i < n) ws[i] = 0.0f;
}

// ---------------------------------------------------------------------------
// Kernel 1: streaming segmented sum.
// Each block: one batch, 4096 contiguous pixels. Index loaded ONCE per pixel
// (int4, coalesced), kept in registers; then 64 channel streams are read as
// float4 (coalesced b128) and accumulated into an LDS-privatized [ch][sp]
// table via ds_add_f32. Final flush uses global f32 atomics.
// ---------------------------------------------------------------------------
__global__ __launch_bounds__(THREADS) void sp_accum(
    const float* __restrict__ feats,    // [NB, NCH, HW]
    const int*   __restrict__ spix,     // [NB, HW]
    float* __restrict__ g_sums,         // [NB, NSP, NCH]
    float* __restrict__ g_cnts) {       // [NB, NSP]
  __shared__ float lsum[NCH * NSP];     // [c][idx]
  __shared__ float lcnt[NSP];
  const int t     = threadIdx.x;
  const int batch = blockIdx.x / BPB;
  const int blk   = blockIdx.x % BPB;

  for (int e = t; e < NCH * NSP; e += THREADS) lsum[e] = 0.0f;
  if (t < NSP) lcnt[t] = 0.0f;
  __syncthreads();

  const long pbase = (long)blk * PIX_PER_BLK;
  const int* idxp  = spix + (long)batch * HW + pbase;

  // Load 16 pixel indices per thread (4 x int4), fully coalesced.
  int4 idx[4];
#pragma unroll
  for (int g = 0; g < 4; ++g)
    idx[g] = *(const int4*)(idxp + g * 1024 + t * 4);

  // Counts: one pass only.
#pragma unroll
  for (int g = 0; g < 4; ++g) {
    atomicAdd(&lcnt[idx[g].x], 1.0f);
    atomicAdd(&lcnt[idx[g].y], 1.0f);
    atomicAdd(&lcnt[idx[g].z], 1.0f);
    atomicAdd(&lcnt[idx[g].w], 1.0f);
  }

  // Channel streams: contiguous per channel, stride HW between channels.
  for (int c = 0; c < NCH; ++c) {
    const float* fc = feats + (long)(batch * NCH + c) * HW + pbase;
    if (c + 1 < NCH) {
      // prefetch next channel's 16KB chunk (256 threads x 64B spacing)
      __builtin_prefetch(fc + HW + t * 16, 0, 0);
    }
    float* row = &lsum[c * NSP];
#pragma unroll
    for (int g = 0; g < 4; ++g) {
      float4 fv = *(const float4*)(fc + g * 1024 + t * 4);
      atomicAdd(&row[idx[g].x], fv.x);
      atomicAdd(&row[idx[g].y], fv.y);
      atomicAdd(&row[idx[g].z], fv.z);
      atomicAdd(&row[idx[g].w], fv.w);
    }
  }
  __syncthreads();

  // Flush LDS [c][i] -> global [i][c]
  for (int e = t; e < NCH * NSP; e += THREADS) {
    int c = e / NSP, i = e % NSP;
    atomicAdd(&g_sums[(long)batch * NSP * NCH + i * NCH + c], lsum[e]);
  }
  if (t < NSP) atomicAdd(&g_cnts[batch * NSP + t], lcnt[t]);
}

// ---------------------------------------------------------------------------
// Kernel 2: finalize means, then sim via fp32 WMMA Gram matrix.
// One block (4 waves) per batch. Each wave computes one 16x16 tile of
// G = sp * sp^T with a chain of 16 V_WMMA_F32_16X16X4_F32 (K=64 total).
// sim[i,j] = 1 + G[i,j] - 0.5*(|sp_i|^2 + |sp_j|^2).
// ---------------------------------------------------------------------------
__global__ __launch_bounds__(128) void sp_finalize(
    const float* __restrict__ g_sums,   // [NB, NSP, NCH]
    const float* __restrict__ g_cnts,   // [NB, NSP]
    float* __restrict__ out_sp,         // [NB, NSP, NCH]
    float* __restrict__ out_sim) {      // [NB, NSP, NSP]
  __shared__ float sp[NSP * NCH];
  __shared__ float nrm[NSP];
  const int t     = threadIdx.x;
  const int batch = blockIdx.x;

  // means -> LDS and out_sp
  for (int e = t; e < NSP * NCH; e += 128) {
    int i    = e >> 6;
    float cn = g_cnts[batch * NSP + i];
    float s  = g_sums[(long)batch * NSP * NCH + e];
    float v  = (cn > 0.0f) ? (s / cn) : 0.0f;
    sp[e] = v;
    out_sp[(long)batch * NSP * NCH + e] = v;
  }
  __syncthreads();

  // row norms
  if (t < NSP) {
    float acc = 0.0f;
    for (int c = 0; c < NCH; ++c) { float x = sp[t * NCH + c]; acc += x * x; }
    nrm[t] = acc;
  }
  __syncthreads();

  // 4 waves -> 4 tiles of the 32x32 Gram matrix
  const int wave = t >> 5;             // 0..3
  const int lane = t & 31;
  const int It   = (wave >> 1) * 16;   // row-tile base
  const int Jt   = (wave & 1) * 16;    // col-tile base
  const int hf   = lane >> 4;          // lane half (K split)
  const int m    = lane & 15;          // row within tile

  // A fragment (16x4 f32): lane holds sp[It+m, kb+2*hf+{0,1}]
  // B fragment (4x16 f32): identical gather on the J-tile rows (B striped like C/D)
  const float* rowA = &sp[(It + m) * NCH + 2 * hf];
  const float* rowB = &sp[(Jt + m) * NCH + 2 * hf];

  v8f acc = {};
#pragma unroll
  for (int kk = 0; kk < 16; ++kk) {
    v2f a = *(const v2f*)(rowA + kk * 4);
    v2f b = *(const v2f*)(rowB + kk * 4);
    // (neg_a, A, neg_b, B, c_mod, C, reuse_a, reuse_b)
    acc = __builtin_amdgcn_wmma_f32_16x16x4_f32(
        false, a, false, b, (short)0, acc, false, false);
  }

  // C/D layout: VGPR r -> M = r + 8*half, N = lane&15
  const int j = Jt + m;
#pragma unroll
  for (int r = 0; r < 8; ++r) {
    int i = It + r + hf * 8;
    out_sim[batch * NSP * NSP + i * NSP + j] =
        1.0f + acc[r] - 0.5f * (nrm[i] + nrm[j]);
  }
}

// ---------------------------------------------------------------------------
extern "C" void kernel_launch(void* const* d_in, const int* in_sizes, int n_in,
                              void* d_out, int out_size, void* d_ws, size_t ws_size,
                              hipStream_t stream) {
  const float* feats = (const float*)d_in[0];   // [16,64,512,512] f32
  const int*   spix  = (const int*)d_in[1];     // [16,512,512] i32
  float* out    = (float*)d_out;                // sp (32768) ++ sim (16384)
  float* g_sums = (float*)d_ws;                 // [NB*NSP*NCH]
  float* g_cnts = g_sums + NB * NSP * NCH;      // [NB*NSP]

  int nz = NB * NSP * NCH + NB * NSP;
  zero_ws<<<(nz + 255) / 256, 256, 0, stream>>>(g_sums, nz);
  sp_accum<<<NB * BPB, THREADS, 0, stream>>>(feats, spix, g_sums, g_cnts);
  sp_finalize<<<NB, 128, 0, stream>>>(g_sums, g_cnts, out, out + NB * NSP * NCH);
}